// SANBet_Layer_24730421690890
// MI455X (gfx1250) — compile-verified
//
#include <hip/hip_runtime.h>
#include <stdint.h>

typedef __attribute__((ext_vector_type(2))) float v2f;
typedef __attribute__((ext_vector_type(8))) float v8f;

#define D_FEAT 128
#define ROWS_PER_TILE 16
#define CHUNK 32
#define WAVES_PER_BLOCK 4

// ---- CDNA5 async global->LDS staging (ASYNCcnt-tracked) ----
__device__ __forceinline__ void async_load_b32(uint32_t lds_off, const void* gptr) {
    asm volatile("global_load_async_to_lds_b32 %0, %1, off"
                 :: "v"(lds_off), "v"((uint64_t)(uintptr_t)gptr)
                 : "memory");
}
__device__ __forceinline__ void wait_async0() {
    asm volatile("s_wait_asynccnt 0x0" ::: "memory");
}

// uniform binary search: first index i with arr[i] >= key
__device__ __forceinline__ int lower_bound_i(const int* __restrict__ arr, int n, int key) {
    int lo = 0, hi = n;
    while (lo < hi) {
        int mid = (lo + hi) >> 1;
        if (arr[mid] < key) lo = mid + 1; else hi = mid;
    }
    return lo;
}

__global__ __launch_bounds__(WAVES_PER_BLOCK * 32)
void gcn_spmm_wmma(const float* __restrict__ inp,
                   const int*   __restrict__ erow,
                   const int*   __restrict__ ecol,
                   const float* __restrict__ eval,
                   const float* __restrict__ weight,
                   const float* __restrict__ bias,
                   float* __restrict__ out,
                   int n_nodes, int n_edges) {
    // double-buffered per-wave staging: 2 * 4 * 32 * 12B = 3 KB
    __shared__ int   s_row[2][WAVES_PER_BLOCK][CHUNK];
    __shared__ int   s_col[2][WAVES_PER_BLOCK][CHUNK];
    __shared__ float s_val[2][WAVES_PER_BLOCK][CHUNK];

    const int lane = threadIdx.x & 31;
    const int wv   = threadIdx.x >> 5;
    const int tile = blockIdx.x * WAVES_PER_BLOCK + wv;
    const int r0   = tile * ROWS_PER_TILE;
    if (r0 >= n_nodes) return;        // wave-uniform exit

    const int   half = lane >> 4;     // lane half selects K-pair {0,1} vs {2,3}
    const int   n    = lane & 15;     // column index for B/C/D; row index m for A
    const float w    = weight[0];
    const float bi   = bias[0];

    // edge range for rows [r0, r0+16)  (rows are sorted)
    const int estart = lower_bound_i(erow, n_edges, r0);
    const int eend   = lower_bound_i(erow, n_edges, r0 + ROWS_PER_TILE);

    // stage one CHUNK of edges into LDS buffer `buf`; pad tail lanes so the
    // consume loop is branch-free: row=-1 never matches any r0+m -> A=0,
    // col=0 is a safe gather address.
    auto stage = [&](int buf, int cs) {
        const int cnt = eend - cs;    // >= 1 when called; may exceed CHUNK
        if (lane < cnt) {
            const int e = cs + lane;
            async_load_b32((uint32_t)(uintptr_t)&s_row[buf][wv][lane], erow + e);
            async_load_b32((uint32_t)(uintptr_t)&s_col[buf][wv][lane], ecol + e);
            async_load_b32((uint32_t)(uintptr_t)&s_val[buf][wv][lane], eval + e);
        } else {
            s_row[buf][wv][lane] = -1;
            s_col[buf][wv][lane] = 0;
        }
    };

    v8f acc[8] = {};   // 8 feature tiles of 16 cols -> D=128; C[i+8*half][n]

    if (estart < eend) stage(0, estart);

    int buf = 0;
    for (int cs = estart; cs < eend; cs += CHUNK, buf ^= 1) {
        wait_async0();                         // chunk `buf` resident in LDS
        const int ncs = cs + CHUNK;
        if (ncs < eend) stage(buf ^ 1, ncs);   // overlap next stage w/ compute

        #pragma unroll
        for (int s = 0; s < CHUNK / 4; ++s) {  // 4 edges per WMMA K=4 step
            const int lk0 = (s << 2) + (half << 1);  // this lane-half's K-pair
            const int   ra = s_row[buf][wv][lk0];
            const int   rb = s_row[buf][wv][lk0 + 1];
            const int   c0 = s_col[buf][wv][lk0];
            const int   c1 = s_col[buf][wv][lk0 + 1];
            const float v0 = s_val[buf][wv][lk0];
            const float v1 = s_val[buf][wv][lk0 + 1];

            v2f a;                            // A[m][k], m = lane&15
            a.x = (ra - r0 == n) ? w * v0 : 0.0f;
            a.y = (rb - r0 == n) ? w * v1 : 0.0f;

            // issue all 16 B gathers before the WMMA chain (pipelinable)
            const float* rp0 = inp + (size_t)c0 * D_FEAT + n;
            const float* rp1 = inp + (size_t)c1 * D_FEAT + n;
            float bx[8], by[8];
            #pragma unroll
            for (int f = 0; f < 8; ++f) {
                bx[f] = rp0[f * 16];
                by[f] = rp1[f * 16];
            }
            #pragma unroll
            for (int f = 0; f < 8; ++f) {
                v2f b; b.x = bx[f]; b.y = by[f];   // B[k][n] = inp[col_k][f*16+n]
                acc[f] = __builtin_amdgcn_wmma_f32_16x16x4_f32(
                    /*neg_a=*/false, a, /*neg_b=*/false, b,
                    /*c_mod=*/(short)0, acc[f],
                    /*reuse_a=*/false, /*reuse_b=*/false);
            }
        }
    }

    // write D tile + bias; each element owned exclusively by this wave
    #pragma unroll
    for (int f = 0; f < 8; ++f) {
        #pragma unroll
        for (int i = 0; i < 8; ++i) {
            const int row = r0 + i + (half << 3);
            if (row < n_nodes)
                out[(size_t)row * D_FEAT + f * 16 + n] = acc[f][i] + bi;
        }
    }
}

extern "C" void kernel_launch(void* const* d_in, const int* in_sizes, int n_in,
                              void* d_out, int out_size, void* d_ws, size_t ws_size,
                              hipStream_t stream) {
    const float* inp    = (const float*)d_in[0];
    const int*   erow   = (const int*)  d_in[1];
    const int*   ecol   = (const int*)  d_in[2];
    const float* eval   = (const float*)d_in[3];
    const float* weight = (const float*)d_in[4];
    const float* bias   = (const float*)d_in[5];
    float*       out    = (float*)d_out;

    const int n_edges = in_sizes[1];
    const int n_nodes = in_sizes[0] / D_FEAT;

    const int tiles  = (n_nodes + ROWS_PER_TILE - 1) / ROWS_PER_TILE;
    const int blocks = (tiles + WAVES_PER_BLOCK - 1) / WAVES_PER_BLOCK;

    gcn_spmm_wmma<<<blocks, WAVES_PER_BLOCK * 32, 0, stream>>>(
        inp, erow, ecol, eval, weight, bias, out, n_nodes, n_edges);
}